// AttentionLayer_3D_66889820668333
// MI455X (gfx1250) — compile-verified
//
#include <hip/hip_runtime.h>
#include <math.h>

#define TDIM 32
#define SDIM 2048
#define BDIM 512
#define BT   16               // batch tile per workgroup
#define SB   32               // s positions staged per iteration
#define STAGES 4              // s per workgroup = SB*STAGES = 128
#define SCHUNK (SB*STAGES)
#define NSY  (SDIM / SCHUNK)  // 16 s-chunks
#define MSTRIDE 1089          // 33*33 padded LDS strides (bank-conflict aware)
#define TSTRIDE 33
#define XSZ (BT*MSTRIDE)      // floats per LDS tile (17424)

typedef __attribute__((ext_vector_type(2))) float v2f;
typedef __attribute__((ext_vector_type(8))) float v8f;

// ---- CDNA5 async global->LDS copy (ASYNCcnt-tracked), with safe fallback ----
__device__ __forceinline__ void async_g2l_f32(const float* g, float* l) {
#if __has_builtin(__builtin_amdgcn_global_load_async_to_lds_b32)
  __builtin_amdgcn_global_load_async_to_lds_b32(
      (__attribute__((address_space(1))) int*)(void*)g,
      (__attribute__((address_space(3))) int*)(void*)l, 0, 0);
#else
  *l = *g;
#endif
}

__device__ __forceinline__ void wait_async() {
#if __has_builtin(__builtin_amdgcn_s_wait_asynccnt)
  __builtin_amdgcn_s_wait_asynccnt(0);
#else
  asm volatile("s_wait_asynccnt 0" ::: "memory");
#endif
}

// grid = (BDIM/BT, NSY), block = 128 (4 waves). Each wave computes 8 s per stage.
extern "C" __global__ __launch_bounds__(128)
void attn_main(const float* __restrict__ inp, const float* __restrict__ mask,
               const float* __restrict__ W, const float* __restrict__ bias,
               float* __restrict__ ws) {
  extern __shared__ float smem[];
  float* sx = smem;         // raw inputs tile  [m][t][s] padded
  float* sm = smem + XSZ;   // raw mask tile

  const int tid  = threadIdx.x;
  const int lane = tid & 31;
  const int wave = tid >> 5;
  const int li   = lane & 15;
  const int half = lane >> 4;

  const int b0      = blockIdx.x * BT;
  const int sy      = blockIdx.y;
  const int schunk0 = sy * SCHUNK;

  float oacc0[8] = {0,0,0,0,0,0,0,0};   // C-layout output accumulators, u=0..15
  float oacc1[8] = {0,0,0,0,0,0,0,0};   // u=16..31

  for (int st = 0; st < STAGES; ++st) {
    const int s0 = schunk0 + st * SB;

    // ---- stage: async copy 16x32x32 x/mask tile, lanes along s (contiguous) ----
    for (int row = wave; row < BT * TDIM; row += 4) {
      const int m = row >> 5, t = row & 31;
      const size_t goff = ((size_t)(b0 + m) * TDIM + t) * SDIM + s0 + lane;
      const int    loff = m * MSTRIDE + t * TSTRIDE + lane;
      async_g2l_f32(inp  + goff, sx + loff);
      async_g2l_f32(mask + goff, sm + loff);
    }
    wait_async();
    __syncthreads();

    // ---- compute: 8 s positions per wave ----
    for (int j = 0; j < 8; ++j) {
      const int sl = wave * 8 + j;
      const int s  = s0 + sl;

      // A fragments (x = inputs*mask), f32 16x4 layout: lane holds M=li, K=4c+2*half(+1)
      v2f a[8];
      #pragma unroll
      for (int c = 0; c < 8; ++c) {
        const int k  = 4 * c + 2 * half;
        const int i0 = li * MSTRIDE + k * TSTRIDE + sl;
        a[c].x = sx[i0]           * sm[i0];
        a[c].y = sx[i0 + TSTRIDE] * sm[i0 + TSTRIDE];
      }

      // B fragments from W[s] (32x32, row-major t,u) and 16 fp32 WMMAs
      const float* Wp = W + (size_t)s * (TDIM * TDIM);
      v8f z0 = {}; v8f z1 = {};
      #pragma unroll
      for (int c = 0; c < 8; ++c) {
        const int k = 4 * c + 2 * half;
        v2f bf0, bf1;
        bf0.x = Wp[k * TDIM + li];          bf0.y = Wp[(k + 1) * TDIM + li];
        bf1.x = Wp[k * TDIM + 16 + li];     bf1.y = Wp[(k + 1) * TDIM + 16 + li];
        z0 = __builtin_amdgcn_wmma_f32_16x16x4_f32(false, a[c], false, bf0,
                                                   (short)0, z0, false, false);
        z1 = __builtin_amdgcn_wmma_f32_16x16x4_f32(false, a[c], false, bf1,
                                                   (short)0, z1, false, false);
      }
      const float bv0 = bias[s * TDIM + li];
      const float bv1 = bias[s * TDIM + 16 + li];

      // e = exp(tanh(z + b)); row-sum over u via intra-16-lane xor shuffles
      float e0[8], e1[8], rs[8];
      #pragma unroll
      for (int r = 0; r < 8; ++r) {
        e0[r] = expf(tanhf(z0[r] + bv0));
        e1[r] = expf(tanhf(z1[r] + bv1));
        float t2 = e0[r] + e1[r];
        t2 += __shfl_xor(t2, 1);
        t2 += __shfl_xor(t2, 2);
        t2 += __shfl_xor(t2, 4);
        t2 += __shfl_xor(t2, 8);
        rs[r] = 1.0f / (t2 + 1e-7f);
      }

      // out += att * x ; x re-read from LDS in C layout (lane = u, rows in regs)
      #pragma unroll
      for (int r = 0; r < 8; ++r) {
        const int m  = r + 8 * half;
        const int ia = m * MSTRIDE + li * TSTRIDE + sl;   // t = li
        const int ib = ia + 16 * TSTRIDE;                 // t = li + 16
        const float xc0 = sx[ia] * sm[ia];
        const float xc1 = sx[ib] * sm[ib];
        oacc0[r] += e0[r] * rs[r] * xc0;
        oacc1[r] += e1[r] * rs[r] * xc1;
      }
    }
    __syncthreads();   // protect LDS tile before next stage restages it
  }

  // ---- cross-wave reduction through LDS, one partial tile per workgroup ----
  #pragma unroll
  for (int v = 0; v < 8; ++v) {
    sx[wave * 512 + v * 32 + lane]       = oacc0[v];
    sx[wave * 512 + (v + 8) * 32 + lane] = oacc1[v];
  }
  __syncthreads();
  if (wave == 0) {
    #pragma unroll
    for (int v = 0; v < 16; ++v) {
      float sum = 0.f;
      #pragma unroll
      for (int w = 0; w < 4; ++w) sum += sx[w * 512 + v * 32 + lane];
      const int nt = v >> 3, r = v & 7;
      const int m  = r + 8 * half;
      const int t  = nt * 16 + li;
      ws[(size_t)sy * (BDIM * TDIM) + (b0 + m) * TDIM + t] = sum;
    }
  }
}

// deterministic final reduction over the NSY s-chunk partials
extern "C" __global__ void attn_reduce(const float* __restrict__ ws,
                                       float* __restrict__ out) {
  const int idx = blockIdx.x * blockDim.x + threadIdx.x;
  if (idx < BDIM * TDIM) {
    float s = 0.f;
    #pragma unroll
    for (int c = 0; c < NSY; ++c) s += ws[(size_t)c * (BDIM * TDIM) + idx];
    out[idx] = s;
  }
}

extern "C" void kernel_launch(void* const* d_in, const int* in_sizes, int n_in,
                              void* d_out, int out_size, void* d_ws, size_t ws_size,
                              hipStream_t stream) {
  (void)in_sizes; (void)n_in; (void)out_size; (void)ws_size;
  const float* inp  = (const float*)d_in[0];
  const float* mask = (const float*)d_in[1];
  const float* W    = (const float*)d_in[2];
  const float* bias = (const float*)d_in[3];
  float* ws  = (float*)d_ws;   // NSY * 512 * 32 floats = 1 MB
  float* out = (float*)d_out;

  dim3 grid(BDIM / BT, NSY);
  const size_t shmem = (size_t)2 * XSZ * sizeof(float);  // ~136 KB of 320 KB/WGP
  attn_main<<<grid, 128, shmem, stream>>>(inp, mask, W, bias, ws);
  attn_reduce<<<(BDIM * TDIM + 255) / 256, 256, 0, stream>>>(ws, out);
}